// YoloDistLoss_57088705298621
// MI455X (gfx1250) — compile-verified
//
#include <hip/hip_runtime.h>
#include <math.h>

// ---------------- problem constants (from reference) ----------------
#define NB 16
#define NT 50
#define NA 5
#define NH 128
#define NW 128
#define NC 20
#define NCELL (NB * NA * NH * NW)      // 1,310,720
#define PRED_STRIDE (6 + NC)           // 26
#define TGT_STRIDE (13 + NC)           // 33
#define SCALE_F 16.0f
#define PI_F 3.14159265358979323846f
#define BAD_CONF_W 1.25f

// k_conf launch shape: trip count is uniform (grid*block divides NCELL twice)
#define CONF_BLOCKS 2560
#define CONF_THREADS 256

typedef float v2f __attribute__((ext_vector_type(2)));
typedef float v8f __attribute__((ext_vector_type(8)));

// -------- CDNA5 feature detection (compile-safe on host pass too) --------
#if defined(__HIP_DEVICE_COMPILE__) && __has_builtin(__builtin_amdgcn_wmma_f32_16x16x4_f32)
#define HAVE_WMMA_F32X4 1
#endif

#if defined(__HIP_DEVICE_COMPILE__) &&                                   \
    __has_builtin(__builtin_amdgcn_global_load_async_to_lds_b32) &&      \
    __has_builtin(__builtin_amdgcn_s_wait_asynccnt)
#define HAVE_ASYNC_LDS 1
// b32 variant takes typed int* pointers: (global int*, local int*, imm, imm)
typedef __attribute__((address_space(1))) int* as1_ip;
typedef __attribute__((address_space(3))) int* as3_ip;
#endif

// ---------------- helpers ----------------
__device__ __forceinline__ float inv_tanh_f(float y) {
  float ys = fminf(fmaxf(y, -1.0f + 1e-6f), 1.0f - 1e-6f);
  float at = 0.5f * logf((1.0f + ys) / (1.0f - ys));
  return (y <= -1.0f) ? -2.0f : ((y >= 1.0f) ? 2.0f : at);
}

// 32-lane sum on the matrix unit: A(16x4) holds per-lane partials
// (lane l<16 -> A[l][0], lane l>=16 -> A[l-16][2]); B = ones, so
// D[m][n] = p(m) + p(m+16). Summing a lane's 8 D VGPRs covers rows 0..7
// (lanes 0-15) or 8..15 (lanes 16-31); one shfl_xor(16) completes.
// EXEC must be all ones at the call site (WMMA requirement).
__device__ __forceinline__ float wave_sum(float p) {
#if defined(HAVE_WMMA_F32X4)
  v2f a; a[0] = p;    a[1] = 0.0f;
  v2f b; b[0] = 1.0f; b[1] = 1.0f;
  v8f c = {};
  c = __builtin_amdgcn_wmma_f32_16x16x4_f32(
      /*neg_a=*/false, a, /*neg_b=*/false, b,
      /*c_mod=*/(short)0, c, /*reuse_a=*/false, /*reuse_b=*/false);
  float s = ((c[0] + c[1]) + (c[2] + c[3])) + ((c[4] + c[5]) + (c[6] + c[7]));
  s += __shfl_xor(s, 16, 32);
  return s;
#else
  for (int off = 16; off > 0; off >>= 1) p += __shfl_xor(p, off, 32);
  return p;
#endif
}

__device__ __forceinline__ int cell_of(int b, int a, int gj, int gi) {
  return ((b * NA + a) * NH + gj) * NW + gi;
}

// ---------------- kernels ----------------

// Zero/one-init scratch: conf_mask=1, winner=-1, acc[0..7]=0.
__global__ void k_init(float* __restrict__ conf_mask, int* __restrict__ winner,
                       float* __restrict__ acc) {
  int i = blockIdx.x * blockDim.x + threadIdx.x;
  if (i < NCELL) {
    conf_mask[i] = 1.0f;
    winner[i] = -1;
  }
  if (i < 8) acc[i] = 0.0f;
}

// Per (b,t): anchor distances, ignore-scatter (conf_mask=0), winner scatter.
__global__ void k_targets(const float* __restrict__ tgt,
                          const int* __restrict__ tsz,
                          const float* __restrict__ anchors,
                          float* __restrict__ conf_mask,
                          int* __restrict__ winner,
                          int* __restrict__ packed) {
  int tid = blockIdx.x * blockDim.x + threadIdx.x;
  if (tid >= NB * NT) return;
  int b = tid / NT, t = tid % NT;
  const float* row = tgt + (size_t)tid * TGT_STRIDE;
  float gx = row[0] / SCALE_F, gy = row[1] / SCALE_F;
  float gh = row[3] / SCALE_F, gw = row[4] / SCALE_F;
  bool valid = (t < tsz[b]) && (gw != 0.0f) && (gh != 0.0f);
  int gi = min(max((int)gx, 0), NW - 1);
  int gj = min(max((int)gy, 0), NH - 1);

  float cpts[8];
#pragma unroll
  for (int i = 0; i < 8; i++)
    cpts[i] = row[5 + i] / SCALE_F - (((i & 1) == 0) ? gx : gy);

  int best = 0;
  float bestd = 3.4e38f;
#pragma unroll
  for (int a = 0; a < NA; a++) {
    float aw = anchors[a * 3 + 0] / SCALE_F;
    float ah = anchors[a * 3 + 1] / SCALE_F;
    float ar = anchors[a * 3 + 2];
    float cr = cosf(ar), sr = sinf(ar);
    float ap[8] = {-cr * aw, sr * aw,  cr * aw, -sr * aw,
                   -sr * ah, -cr * ah, sr * ah,  cr * ah};
    float dn = 0.0f;
#pragma unroll
    for (int i = 0; i < 8; i += 2) {
      float dx = cpts[i] - ap[i], dy = cpts[i + 1] - ap[i + 1];
      dn += sqrtf(dx * dx + dy * dy);
    }
    float hw = (ah + aw) * 0.5f;
    float nrm = ((gh + gw) * 0.5f + hw) * 0.5f;
    float q = dn / nrm;
    float dist = q * q;
    if (dist < bestd) { bestd = dist; best = a; }      // first-min (argmin)
    if (valid && dist < 0.5f)                           // IGNORE_THRES
      conf_mask[cell_of(b, a, gj, gi)] = 0.0f;          // min-with-0 == store 0
  }

  if (valid) {
    atomicMax(&winner[cell_of(b, best, gj, gi)], t);    // dedupe duplicate idx
    packed[tid] = (best << 16) | (gj << 8) | gi;
  } else {
    packed[tid] = -1;
  }
}

// After the ignore pass: conf_mask=1 at every valid target cell (ref ordering).
__global__ void k_set_one(const int* __restrict__ packed,
                          float* __restrict__ conf_mask) {
  int tid = blockIdx.x * blockDim.x + threadIdx.x;
  if (tid >= NB * NT) return;
  int p = packed[tid];
  if (p < 0) return;
  int b = tid / NT;
  conf_mask[cell_of(b, (p >> 16) & 0xFF, (p >> 8) & 0xFF, p & 0xFF)] = 1.0f;
}

// Streaming conf-BCE over all cells. The strided pred[...,0] gather is
// double-buffered through LDS with the async data mover (issue i+1, wait<=1,
// consume i) so the transfer overlaps the BCE math and the coalesced
// conf_mask/winner loads. Block reduction finishes on the WMMA unit.
__global__ void __launch_bounds__(CONF_THREADS)
k_conf(const float* __restrict__ pred, const float* __restrict__ conf_mask,
       const int* __restrict__ winner, float* __restrict__ acc) {
#if defined(HAVE_ASYNC_LDS)
  __shared__ float stage[2][CONF_THREADS];
#endif
  __shared__ float wsum[8 * 4];

  float p0 = 0.f, p1 = 0.f, p2 = 0.f, p3 = 0.f;
  const int stride = gridDim.x * blockDim.x;
  const int idx0 = blockIdx.x * blockDim.x + threadIdx.x;

#if defined(HAVE_ASYNC_LDS)
  // Prologue: stage iteration 0's conf value. Trip count is uniform across
  // the whole grid (stride divides NCELL), so all control flow below is
  // wave-uniform and EXEC stays all-ones.
  if (idx0 < NCELL) {
    __builtin_amdgcn_global_load_async_to_lds_b32(
        (as1_ip)(void*)(pred + (size_t)idx0 * PRED_STRIDE),
        (as3_ip)(void*)&stage[0][threadIdx.x], 0, 0);
  }
  int buf = 0;
#endif

  for (int cell = idx0; cell < NCELL; cell += stride) {
    float cm = conf_mask[cell];
    float m = (winner[cell] >= 0) ? 1.0f : 0.0f;  // mask == tconf
    float z;
#if defined(HAVE_ASYNC_LDS)
    int next = cell + stride;
    if (next < NCELL) {
      // Kick off the next tile before consuming this one.
      __builtin_amdgcn_global_load_async_to_lds_b32(
          (as1_ip)(void*)(pred + (size_t)next * PRED_STRIDE),
          (as3_ip)(void*)&stage[buf ^ 1][threadIdx.x], 0, 0);
      __builtin_amdgcn_s_wait_asynccnt(1);  // oldest (this iter's) is done
    } else {
      __builtin_amdgcn_s_wait_asynccnt(0);  // drain on the last iteration
    }
    z = stage[buf][threadIdx.x];
    buf ^= 1;
#else
    const float* gp = pred + (size_t)cell * PRED_STRIDE;
    __builtin_prefetch(gp + (size_t)stride * PRED_STRIDE, 0, 0);
    z = *gp;
#endif
    float cmf = cm * (1.0f - m);  // conf_mask_false
    // stable BCE-with-logits, target = m
    float l = fmaxf(z, 0.0f) - z * m + log1pf(expf(-fabsf(z)));
    p0 += l * cmf;  // numerator of false-term
    p1 += cmf;      // denominator of false-term
    p2 += l * m;    // numerator of true-term
    p3 += m;        // denominator of true-term
  }

  // All lanes reach here uniformly -> EXEC all ones -> WMMA-legal.
  p0 = wave_sum(p0);
  p1 = wave_sum(p1);
  p2 = wave_sum(p2);
  p3 = wave_sum(p3);

  int lane = threadIdx.x & 31, wv = threadIdx.x >> 5;
  if (lane == 0) {
    wsum[wv * 4 + 0] = p0;
    wsum[wv * 4 + 1] = p1;
    wsum[wv * 4 + 2] = p2;
    wsum[wv * 4 + 3] = p3;
  }
  __syncthreads();
  if (threadIdx.x == 0) {
    float a0 = 0.f, a1 = 0.f, a2 = 0.f, a3 = 0.f;
    for (int i = 0; i < 8; i++) {
      a0 += wsum[i * 4 + 0];
      a1 += wsum[i * 4 + 1];
      a2 += wsum[i * 4 + 2];
      a3 += wsum[i * 4 + 3];
    }
    atomicAdd(&acc[0], a0);
    atomicAdd(&acc[1], a1);
    atomicAdd(&acc[2], a2);
    atomicAdd(&acc[3], a3);
  }
}

// Sparse masked losses at winner cells only (<= 800 threads do work).
__global__ void k_masked(const float* __restrict__ pred,
                         const float* __restrict__ tgt,
                         const float* __restrict__ anchors,
                         const int* __restrict__ winner,
                         const int* __restrict__ packed,
                         float* __restrict__ acc) {
  int tid = blockIdx.x * blockDim.x + threadIdx.x;
  if (tid >= NB * NT) return;
  int p = packed[tid];
  if (p < 0) return;
  int b = tid / NT, t = tid % NT;
  int best = (p >> 16) & 0xFF, gj = (p >> 8) & 0xFF, gi = p & 0xFF;
  int cell = cell_of(b, best, gj, gi);
  if (winner[cell] != t) return;  // one accounting per masked cell

  const float* row = tgt + (size_t)tid * TGT_STRIDE;
  float gx = row[0] / SCALE_F, gy = row[1] / SCALE_F, gr = row[2];
  float gh = row[3] / SCALE_F, gw = row[4] / SCALE_F;
  float aw = anchors[best * 3 + 0] / SCALE_F;
  float ah = anchors[best * 3 + 1] / SCALE_F;
  float ar = anchors[best * 3 + 2];

  float tx = inv_tanh_f(gx - ((float)gi + 0.5f));
  float ty = inv_tanh_f(gy - ((float)gj + 0.5f));
  float rd = gr - ar;
  rd = (rd > PI_F) ? rd - 2.0f * PI_F : ((rd < -PI_F) ? rd + 2.0f * PI_F : rd);
  float tr = inv_tanh_f(rd / (PI_F * 0.5f));
  float tw = logf(gw / aw + 1e-16f);
  float th = logf(gh / ah + 1e-16f);

  int tlab = 0;
  float bm = row[13];
  for (int c = 1; c < NC; c++) {
    float v = row[13 + c];
    if (v > bm) { bm = v; tlab = c; }
  }

  const float* pb = pred + (size_t)cell * PRED_STRIDE;
  float x = pb[1], y = pb[2], r = pb[3], h = pb[4], w = pb[5];
  float se = (x - tx) * (x - tx) + (y - ty) * (y - ty) +
             (w - tw) * (w - tw) + (h - th) * (h - th) +
             (r - tr) * (r - tr);

  float mx = pb[6];
  for (int c = 1; c < NC; c++) mx = fmaxf(mx, pb[6 + c]);
  float s = 0.0f;
  for (int c = 0; c < NC; c++) s += expf(pb[6 + c] - mx);
  float lse = mx + logf(s);
  float nll = lse - pb[6 + tlab];

  atomicAdd(&acc[4], se);   // x+y+w+h+r MSE numerators share denominator
  atomicAdd(&acc[5], nll);  // class NLL numerator
  atomicAdd(&acc[6], 1.0f); // sum(mask)
}

__global__ void k_final(const float* __restrict__ acc, float* __restrict__ out) {
  if (blockIdx.x == 0 && threadIdx.x == 0) {
    float loss_conf = BAD_CONF_W * acc[0] / fmaxf(acc[1], 1.0f) +
                      acc[2] / fmaxf(acc[3], 1.0f);
    float denom = fmaxf(acc[6], 1.0f);
    out[0] = acc[4] / denom + acc[5] / denom + loss_conf;
  }
}

// ---------------- host launcher ----------------
extern "C" void kernel_launch(void* const* d_in, const int* in_sizes, int n_in,
                              void* d_out, int out_size, void* d_ws,
                              size_t ws_size, hipStream_t stream) {
  (void)in_sizes; (void)n_in; (void)out_size; (void)ws_size;
  const float* pred    = (const float*)d_in[0];
  const float* tgt     = (const float*)d_in[1];
  const int*   tsz     = (const int*)d_in[2];
  const float* anchors = (const float*)d_in[3];
  float* out = (float*)d_out;

  char* ws = (char*)d_ws;
  float* conf_mask = (float*)ws;                                   // 5.24 MB
  int*   winner    = (int*)(ws + sizeof(float) * NCELL);           // 5.24 MB
  float* acc       = (float*)(ws + (sizeof(float) + sizeof(int)) * NCELL);
  int*   packed    = (int*)((char*)acc + 64);                      // 3.2 KB

  k_init<<<(NCELL + 255) / 256, 256, 0, stream>>>(conf_mask, winner, acc);
  k_targets<<<(NB * NT + 255) / 256, 256, 0, stream>>>(tgt, tsz, anchors,
                                                       conf_mask, winner, packed);
  k_set_one<<<(NB * NT + 255) / 256, 256, 0, stream>>>(packed, conf_mask);
  k_conf<<<CONF_BLOCKS, CONF_THREADS, 0, stream>>>(pred, conf_mask, winner, acc);
  k_masked<<<(NB * NT + 255) / 256, 256, 0, stream>>>(pred, tgt, anchors,
                                                      winner, packed, acc);
  k_final<<<1, 1, 0, stream>>>(acc, out);
}